// PrototypeClassifier_27169963114958
// MI455X (gfx1250) — compile-verified
//
#include <hip/hip_runtime.h>
#include <hip/hip_bf16.h>

typedef float v2f __attribute__((ext_vector_type(2)));
typedef float v8f __attribute__((ext_vector_type(8)));
typedef unsigned int u32x4 __attribute__((ext_vector_type(4)));
typedef int i32x4 __attribute__((ext_vector_type(4)));
typedef int i32x8 __attribute__((ext_vector_type(8)));

// Problem sizes (fixed by the reference)
constexpr int BATCH = 1024;
constexpr int NCLS  = 100;
constexpr int NPROT = 10;   // padded to 16 (WMMA N dim)
constexpr int DIM   = 512;

constexpr int KC      = 64;       // K chunk staged in LDS (TDM pad_interval=64 DW)
constexpr int PAD     = 4;        // LDS row padding in DWORDs (TDM pad_amount=4 DW)
constexpr int KSTRIDE = KC + PAD; // padded row stride
constexpr int WAVES   = 8;        // waves per block -> 8 batch tiles of 16 rows
constexpr int MROWS   = WAVES * 16;  // 128 batch rows per block

static_assert(KC == 64 && PAD == 4, "TDM descriptor pad encoding assumes 64+4");

// Issue one Tensor-DMA: copy a (MROWS x KC) fp32 tile of x into LDS,
// inserting PAD dwords of padding after every KC dwords (row stride KSTRIDE).
__device__ __forceinline__ void tdm_load_x_tile(unsigned lds_byte_off,
                                                unsigned long long gaddr) {
    u32x4 g0;
    g0.x = 1u;                                         // count=1 (valid), user mode
    g0.y = lds_byte_off;                               // LDS destination (bytes)
    g0.z = (unsigned)(gaddr & 0xFFFFFFFFu);            // global_addr[31:0]
    g0.w = (unsigned)((gaddr >> 32) & 0x1FFFFFFu)      // global_addr[56:32]
         | 0x80000000u;                                // type=2 ("image")

    i32x8 g1;
    g1[0] = (2 << 16)      // data_size = 4 bytes
          | (1 << 20)      // pad_enable
          | (5 << 22)      // pad_interval: 64 DWORDs between pads
          | (3 << 25);     // pad_amount: 4 DWORDs
    g1[1] = (int)((unsigned)DIM << 16);     // tensor_dim0 = 512 (bits 79:48)
    g1[2] = (int)((unsigned)BATCH << 16);   // tensor_dim1 = 1024 (bits 111:80)
    g1[3] = (int)((unsigned)KC << 16);      // tile_dim0 = 64 (bits 127:112)
    g1[4] = MROWS;                          // tile_dim1 = 128 rows (bits 143:128)
    g1[5] = DIM;                            // tensor_dim0_stride = 512 (bits 207:160)
    g1[6] = 0;                              // stride hi + tensor_dim1_stride lo
    g1[7] = 0;

    i32x4 g2 = {0, 0, 0, 0};                // 2D tile: higher dims unused
    i32x4 g3 = {0, 0, 0, 0};
    i32x8 g4 = {0, 0, 0, 0, 0, 0, 0, 0};    // extra group (zero-filled)

    __builtin_amdgcn_tensor_load_to_lds(g0, g1, g2, g3, g4, /*cpol=*/0);
}

__global__ __launch_bounds__(256)
void PrototypeClassifier_wmma_kernel(const float* __restrict__ x,
                                     const float* __restrict__ proto,
                                     const float* __restrict__ lscales,
                                     float* __restrict__ out) {
    __shared__ float xs [MROWS][KSTRIDE];  // x panel (TDM-filled, hw-padded rows)
    __shared__ float wB [16][KSTRIDE];     // exp(log_scales), padded cols = 0
    __shared__ float wpB[16][KSTRIDE];     // w * proto,       padded cols = 0
    __shared__ float t3s[16];              // sum_d w*p*p  (pad cols = +huge)

    const int c    = blockIdx.x;            // class index
    const int tid  = threadIdx.x;
    const int lane = tid & 31;
    const int wave = tid >> 5;
    const int mb0  = blockIdx.y * MROWS;    // first batch row of this block
    const int mb   = mb0 + wave * 16;       // this wave's batch tile base

    if (tid < 16) t3s[tid] = (tid < NPROT) ? 0.0f : 3.0e38f;
    __syncthreads();

    const int n_lo  = lane & 15;          // row M for A-frag, col N for B-frag
    const int khalf = (lane >> 4) << 1;   // 0 or 2: K sub-offset of this half-wave
    const unsigned xs_lds = (unsigned)(uintptr_t)&xs[0][0];

    v8f acc1 = {};  // t1 = sum_d w * x^2
    v8f acc2 = {};  // t2 = sum_d (w*p) * x

    for (int kc = 0; kc < DIM; kc += KC) {
        // ---- async: TDM pulls the (128 x 64) x panel into LDS (wave 0 only)
        if (wave == 0) {
            const unsigned long long ga =
                (unsigned long long)(uintptr_t)x +
                ((unsigned long long)mb0 * DIM + (unsigned)kc) * 4ull;
            tdm_load_x_tile(xs_lds, ga);
        }

        // ---- meanwhile stage B panels: w = exp(ls), wp = w*p; t3 via LDS atomics
        for (int idx = tid; idx < 16 * KC; idx += 256) {
            const int n = idx >> 6;        // / KC
            const int k = idx & (KC - 1);
            float wv = 0.0f, wpv = 0.0f;
            if (n < NPROT) {
                const int g  = (c * NPROT + n) * DIM + kc + k;
                const float ls = lscales[g];
                const float pv = proto[g];
                wv  = __expf(ls);
                wpv = wv * pv;
                atomicAdd(&t3s[n], wpv * pv);   // ds_add_f32
            }
            wB [n][k] = wv;
            wpB[n][k] = wpv;
        }

        if (wave == 0) __builtin_amdgcn_s_wait_tensorcnt((short)0);
        __syncthreads();   // publish xs + wB/wpB to all waves

        // ---- WMMA K-loop over this chunk (EXEC all-ones, fully uniform)
        const float* xrow  = &xs [wave * 16 + n_lo][khalf];
        const float* wrow  = &wB [n_lo][khalf];
        const float* wprow = &wpB[n_lo][khalf];
#pragma unroll 4
        for (int k = 0; k < KC; k += 4) {
            v2f a   = *(const v2f*)(xrow  + k);   // A frag: x[M][K..K+1]
            v2f bw  = *(const v2f*)(wrow  + k);   // B frag: w^T[K][N]
            v2f bwp = *(const v2f*)(wprow + k);
            v2f asq = a * a;
            acc1 = __builtin_amdgcn_wmma_f32_16x16x4_f32(
                       false, asq, false, bw,  (short)0, acc1, false, false);
            acc2 = __builtin_amdgcn_wmma_f32_16x16x4_f32(
                       false, a,   false, bwp, (short)0, acc2, false, false);
        }
        __syncthreads();   // done reading LDS before next chunk overwrites
    }

    // ---- epilogue: dist = t1 - 2*t2 + t3, then logsumexp over the 16 columns.
    // C/D layout: element r of a lane holds row M = r + 8*(lane>=16), col N = lane&15.
    const float t3v = t3s[n_lo];
    float outv = 0.0f;
#pragma unroll
    for (int r = 0; r < 8; ++r) {
        const float dist = acc1[r] - 2.0f * acc2[r] + t3v;
        float v = -dist;
        float m = v;
#pragma unroll
        for (int mask = 1; mask < 16; mask <<= 1)
            m = fmaxf(m, __shfl_xor(m, mask, 32));
        float e = __expf(v - m);
#pragma unroll
        for (int mask = 1; mask < 16; mask <<= 1)
            e += __shfl_xor(e, mask, 32);
        const float lse = m + __logf(e);
        if (n_lo == r) outv = lse;   // lane r (each half) owns row r / r+8
    }
    if (n_lo < 8) {
        const int row = mb + n_lo + ((lane >> 4) << 3);
        out[row * NCLS + c] = outv;
    }
}

extern "C" void kernel_launch(void* const* d_in, const int* in_sizes, int n_in,
                              void* d_out, int out_size, void* d_ws, size_t ws_size,
                              hipStream_t stream) {
    (void)in_sizes; (void)n_in; (void)out_size; (void)d_ws; (void)ws_size;
    const float* x       = (const float*)d_in[0];
    const float* proto   = (const float*)d_in[1];
    const float* lscales = (const float*)d_in[2];
    float* out           = (float*)d_out;

    dim3 grid(NCLS, BATCH / MROWS);   // (100, 8)
    dim3 block(256);                  // 8 wave32
    PrototypeClassifier_wmma_kernel<<<grid, block, 0, stream>>>(x, proto, lscales, out);
}